// BEADSLayer_3444563771730
// MI455X (gfx1250) — compile-verified
//
#include <hip/hip_runtime.h>
#include <math.h>

#define N_SIG 1024
#define BATCH 32

typedef float v2f __attribute__((ext_vector_type(2)));
typedef float v8f __attribute__((ext_vector_type(8)));

// ---------------------------------------------------------------------------
// Kernel 1: per-(batch,row) band of M = D^T Lambda D + 2*lam0*diag(gamma).
// M is pentadiagonal; store m0 (diag), m1 (off+1), m2 (off+2), stride 4.
// ---------------------------------------------------------------------------
__global__ void beads_mband(const float* __restrict__ x,
                            const float* __restrict__ lg0,
                            const float* __restrict__ lg1,
                            const float* __restrict__ lg2,
                            const float* __restrict__ lgr,
                            float* __restrict__ mb)
{
    int idx = blockIdx.x * blockDim.x + threadIdx.x;
    if (idx >= BATCH * N_SIG) return;
    int b = idx >> 10;
    int k = idx & (N_SIG - 1);
    const int N = N_SIG;
    float lam0 = expf(lg0[0]);
    float lam1 = expf(lg1[0]);
    float lam2 = expf(lg2[0]);
    float r    = expf(lgr[0]);
    const float* xb = x + b * N;

    auto LD1 = [&](int i) -> float {
        if (i < 0 || i > N - 2) return 0.f;
        return lam1 / (fabsf(xb[i + 1] - xb[i]) + 1e-6f);
    };
    auto LD2 = [&](int i) -> float {
        if (i < 0 || i > N - 3) return 0.f;
        return lam2 / (fabsf(xb[i] - 2.f * xb[i + 1] + xb[i + 2]) + 1e-6f);
    };

    float gamma = (1.f + r) / (4.f * fmaxf(fabsf(xb[k]), 1e-6f));
    float l1k  = LD1(k),  l1km  = LD1(k - 1);
    float l2k  = LD2(k),  l2km  = LD2(k - 1), l2km2 = LD2(k - 2);

    mb[idx * 4 + 0] = 2.f * lam0 * gamma + l1km + l1k + l2k + 4.f * l2km + l2km2;
    mb[idx * 4 + 1] = -l1k - 2.f * (l2k + l2km);   // M[k,k+1]
    mb[idx * 4 + 2] = l2k;                         // M[k,k+2]
    mb[idx * 4 + 3] = 0.f;
}

// ---------------------------------------------------------------------------
// Kernel 2: lower band (half-bandwidth 6) of lhs = BTB + A^T M A per batch.
// cb[(b*N+i)*8 + off] = lhs[i+off, i],  off = 0..6 (symmetric), pad slot 7.
// ---------------------------------------------------------------------------
__global__ void beads_lhs_band(const float* __restrict__ A,
                               const float* __restrict__ BTB,
                               const float* __restrict__ mb,
                               float* __restrict__ cb)
{
    int idx = blockIdx.x * blockDim.x + threadIdx.x;
    if (idx >= BATCH * N_SIG) return;
    int b = idx >> 10;
    int i = idx & (N_SIG - 1);
    const int N = N_SIG;
    const float* mbb = mb + (size_t)b * N * 4;

    auto Mval = [&](int k, int l) -> float {
        int d = l - k;
        if (d == 0)  return mbb[k * 4 + 0];
        if (d == 1)  return mbb[k * 4 + 1];
        if (d == -1) return mbb[l * 4 + 1];
        if (d == 2)  return mbb[k * 4 + 2];
        if (d == -2) return mbb[l * 4 + 2];
        return 0.f;
    };

    for (int off = 0; off <= 6; ++off) {
        int j = i + off;
        float acc = 0.f;
        if (j < N) {
            acc = BTB[(size_t)i * N + j];
            for (int k = i - 2; k <= i + 2; ++k) {
                if (k < 0 || k >= N) continue;
                float aki = A[(size_t)k * N + i];
                float ma = 0.f;
                for (int l = k - 2; l <= k + 2; ++l) {
                    if (l < 0 || l >= N) continue;
                    if (l < j - 2 || l > j + 2) continue;  // A[l][j] == 0
                    ma += Mval(k, l) * A[(size_t)l * N + j];
                }
                acc += aki * ma;
            }
        }
        cb[(size_t)idx * 8 + off] = acc;
    }
    cb[(size_t)idx * 8 + 7] = 0.f;
}

// ---------------------------------------------------------------------------
// Kernel 3: one single-wave workgroup per batch. LDL^T factorization of the
// banded SPD lhs in LDS, then forward / diagonal / backward substitution.
// Single-wave workgroup: barriers lower to S_NOP + counters (cheap).
// ---------------------------------------------------------------------------
__global__ void __launch_bounds__(32)
beads_solve(const float* __restrict__ cb_g,
            const float* __restrict__ dvec,
            float* __restrict__ zout)
{
    const int N = N_SIG;
    __shared__ float cb[N_SIG * 8];   // column-major lower band, stride 8
    __shared__ float rv[N_SIG];
    __shared__ float zv[N_SIG];

    int b    = blockIdx.x;
    int lane = threadIdx.x;

    for (int t = lane; t < N * 8; t += 32) cb[t] = cb_g[(size_t)b * N * 8 + t];
    for (int t = lane; t < N; t += 32)     rv[t] = dvec[(size_t)b * N + t];
    __syncthreads();

    // (r,c) pair for the 21 trailing-update entries (1 <= c <= r <= 6)
    int rr = 0, cc = 0;
    if (lane < 21) {
        int t = lane, cnt = 6;
        cc = 1;
        while (t >= cnt) { t -= cnt; ++cc; --cnt; }
        rr = cc + t;
    }

    // ---- LDL^T factorization (right-looking, band stays within 7 rows) ----
    for (int j = 0; j < N; ++j) {
        float dj  = cb[j * 8];
        float inv = 1.f / dj;
        float ar = 0.f, ac = 0.f;
        if (lane < 21) { ar = cb[j * 8 + rr]; ac = cb[j * 8 + cc]; }
        float morig = (lane < 6) ? cb[j * 8 + lane + 1] : 0.f;
        __syncthreads();
        if (lane < 21 && (j + cc) < N)
            cb[(j + cc) * 8 + (rr - cc)] -= (ar * inv) * ac;
        if (lane < 6)
            cb[j * 8 + lane + 1] = morig * inv;   // store multiplier
        __syncthreads();
    }

    // ---- forward solve (unit lower) + diagonal scale: rv[j] <- (L^-1 rhs)_j / d_j
    for (int j = 0; j < N; ++j) {
        float yj = rv[j];
        float mult = 0.f, cur = 0.f;
        bool act = (lane >= 1 && lane <= 6 && (j + lane) < N);
        if (act) { mult = cb[j * 8 + lane]; cur = rv[j + lane]; }
        __syncthreads();
        if (act) rv[j + lane] = cur - mult * yj;
        if (lane == 0) rv[j] = yj / cb[j * 8];
        __syncthreads();
    }

    // ---- backward solve: z_j = v_j - sum_r L[j+r,j] * z_{j+r}  (8-lane reduce)
    for (int j = N - 1; j >= 0; --j) {
        float p = 0.f;
        if (lane >= 1 && lane <= 6 && (j + lane) < N)
            p = -cb[j * 8 + lane] * zv[j + lane];
        if (lane == 0) p = rv[j];
        p += __shfl_xor(p, 1, 8);
        p += __shfl_xor(p, 2, 8);
        p += __shfl_xor(p, 4, 8);
        if (lane == 0) zv[j] = p;
        __syncthreads();
    }

    for (int t = lane; t < N; t += 32) zout[(size_t)b * N + t] = zv[t];
}

// ---------------------------------------------------------------------------
// Kernel 4: x_new = Z @ A^T via V_WMMA_F32_16X16X4_F32 (fp32 matrix pipe).
// One wave per 16x16 output tile; K accumulated 4 at a time.
// ISA layout: A 16x4 -> lane m = lane&15, VGPR v holds K = v + 2*(lane>>4);
//             B 4x16 -> lane c = lane&15, VGPR v holds row K = v + 2*(lane>>4);
//             C/D 16x16 -> VGPR v holds row v + 8*(lane>>4), col lane&15.
// ---------------------------------------------------------------------------
__global__ void __launch_bounds__(32)
beads_out_wmma(const float* __restrict__ Am,
               const float* __restrict__ Z,
               float* __restrict__ out)
{
    const int N = N_SIG;
    int lane = threadIdx.x;
    int mt = blockIdx.x >> 6;        // 2 tiles of batch rows
    int nt = blockIdx.x & 63;        // 64 tiles of output columns
    int h = lane >> 4;
    int c = lane & 15;

    const float* zp = Z  + (size_t)(mt * 16 + c) * N;   // A-fragment: row = batch m
    const float* ap = Am + (size_t)(nt * 16 + c) * N;   // B-fragment: A row = out col

    v8f acc = {0.f, 0.f, 0.f, 0.f, 0.f, 0.f, 0.f, 0.f};
    for (int k = 0; k < N; k += 4) {
        int kk = k + 2 * h;
        v2f a;  a.x  = zp[kk]; a.y  = zp[kk + 1];
        v2f bm; bm.x = ap[kk]; bm.y = ap[kk + 1];
        acc = __builtin_amdgcn_wmma_f32_16x16x4_f32(
                  false, a, false, bm, (short)0, acc, false, false);
    }

    int col = nt * 16 + c;
    #pragma unroll
    for (int v = 0; v < 8; ++v) {
        int row = mt * 16 + v + 8 * h;
        out[(size_t)row * N + col] = acc[v];
    }
}

// ---------------------------------------------------------------------------
extern "C" void kernel_launch(void* const* d_in, const int* in_sizes, int n_in,
                              void* d_out, int out_size, void* d_ws, size_t ws_size,
                              hipStream_t stream)
{
    const float* x    = (const float*)d_in[0];
    // d_in[1] = y (unused by the reference)
    const float* dvec = (const float*)d_in[2];
    const float* A    = (const float*)d_in[3];
    // d_in[4] = D (structure derived analytically)
    const float* BTB  = (const float*)d_in[5];
    const float* lg0  = (const float*)d_in[6];
    const float* lg1  = (const float*)d_in[7];
    const float* lg2  = (const float*)d_in[8];
    const float* lgr  = (const float*)d_in[9];
    float* out = (float*)d_out;

    float* ws = (float*)d_ws;
    float* mb = ws;                                   // 32*1024*4 = 131072 floats
    float* cb = ws + BATCH * N_SIG * 4;               // 32*1024*8 = 262144 floats
    float* z  = ws + BATCH * N_SIG * 4 + BATCH * N_SIG * 8;  // 32768 floats

    dim3 blk256(256), grid128(128);
    beads_mband   <<<grid128, blk256, 0, stream>>>(x, lg0, lg1, lg2, lgr, mb);
    beads_lhs_band<<<grid128, blk256, 0, stream>>>(A, BTB, mb, cb);
    beads_solve   <<<dim3(BATCH), dim3(32), 0, stream>>>(cb, dvec, z);
    beads_out_wmma<<<dim3(128),  dim3(32), 0, stream>>>(A, z, out);
}